// SpherefaceLayer_64055142252613
// MI455X (gfx1250) — compile-verified
//
#include <hip/hip_runtime.h>

typedef __attribute__((ext_vector_type(2))) float v2f;
typedef __attribute__((ext_vector_type(8))) float v8f;

#define NROWS 16384
#define DDIM  128
#define UNITS 8192

#define BLOCK_M 128
#define BLOCK_N 128
#define LDA 132    // A row stride (words): 4*l15 banks -> conflict-free b64 frag reads
#define LDBP 288   // B pair-row stride (words): 288%64==32 -> half-waves use disjoint banks

#define WMMA(acc, a, b) \
    acc = __builtin_amdgcn_wmma_f32_16x16x4_f32(false, a, false, b, (short)0, acc, false, false)

// ---------------------------------------------------------------------------
// Kernel 1: per-column inverse L2 norm of weights [D, U]
// ---------------------------------------------------------------------------
__global__ void colnorm_kernel(const float* __restrict__ W,
                               float* __restrict__ inv_norm) {
    int u = blockIdx.x * blockDim.x + threadIdx.x;
    if (u < UNITS) {
        float s = 0.0f;
#pragma unroll 8
        for (int d = 0; d < DDIM; ++d) {
            float w = W[(size_t)d * UNITS + u];
            s += w * w;
        }
        inv_norm[u] = rsqrtf(fmaxf(s, 1e-12f));
    }
}

// ---------------------------------------------------------------------------
// Kernel 2: out[N,U] = (E[N,D] @ W[D,U]) * inv_norm[u] via V_WMMA_F32_16X16X4_F32
// Block: 256 threads = 8 waves; tile 128(M) x 128(N); full K=128 in LDS.
// Waves: wm in 0..3 (32 rows each), wn in 0..1 (64 cols each).
// Each wave: 2 M-subtiles x 4 N-subtiles = 8 accumulators.
// B stored pair-interleaved: Bs2[pair][col] = {W[2p][c], W[2p+1][c]} so each
// B fragment is a single aligned ds_load_b64 (no VALU packing).
// ---------------------------------------------------------------------------
__global__ __launch_bounds__(256) void sphereface_gemm_kernel(
        const float* __restrict__ E,      // [NROWS, DDIM]
        const float* __restrict__ W,      // [DDIM, UNITS]
        const float* __restrict__ inv_norm,
        float* __restrict__ out)          // [NROWS, UNITS]
{
    __shared__ float As[BLOCK_M * LDA];          // 128 x 128 (stride 132)
    __shared__ float Bs2[(DDIM / 2) * LDBP];     // 64 pair-rows x 256 (stride 288)

    const int tid = threadIdx.x;
    const int m0 = blockIdx.y * BLOCK_M;
    const int n0 = blockIdx.x * BLOCK_N;

    // ---- Stage A tile: 128 rows x 32 float4 (4096 float4, 16 per thread) ----
    const float4* Eg = (const float4*)E;         // row = 32 float4
#pragma unroll
    for (int i = 0; i < 16; ++i) {
        int idx = tid + i * 256;
        int r = idx >> 5, c = idx & 31;
        float4 v = Eg[(size_t)(m0 + r) * (DDIM / 4) + c];
        *(float4*)&As[r * LDA + c * 4] = v;
    }

    // ---- Stage B tile pair-interleaved: 64 pairs x 64 col-pairs ----
    const float2* Wg2 = (const float2*)W;        // row = 4096 float2
#pragma unroll
    for (int i = 0; i < 16; ++i) {
        int idx = tid + i * 256;
        int cp = idx & 63;                       // col-pair (cols 2cp, 2cp+1)
        int p  = idx >> 6;                       // row pair (rows 2p, 2p+1)
        float2 r0v = Wg2[(size_t)(2 * p)     * (UNITS / 2) + (n0 >> 1) + cp];
        float2 r1v = Wg2[(size_t)(2 * p + 1) * (UNITS / 2) + (n0 >> 1) + cp];
        float2* dst = (float2*)&Bs2[p * LDBP + 4 * cp];
        dst[0] = make_float2(r0v.x, r1v.x);      // col 2cp:   {W[2p], W[2p+1]}
        dst[1] = make_float2(r0v.y, r1v.y);      // col 2cp+1: {W[2p], W[2p+1]}
    }
    __syncthreads();

    const int wid  = tid >> 5;
    const int lane = tid & 31;
    const int wm   = wid & 3;                    // 32-row strip
    const int wn   = wid >> 2;                   // 64-col strip
    const int l15  = lane & 15;
    const int hi   = (lane < 16) ? 0 : 1;        // half-wave: K pair select
    const int khalf = hi * 2;

    const int arow0 = wm * 32 + l15;             // A frag row, M-subtile 0
    const int arow1 = arow0 + 16;                // A frag row, M-subtile 1
    const int nloc  = wn * 64 + l15;             // base local column

    v8f acc00 = {}, acc01 = {}, acc02 = {}, acc03 = {};
    v8f acc10 = {}, acc11 = {}, acc12 = {}, acc13 = {};

#pragma unroll 2
    for (int k0 = 0; k0 < DDIM; k0 += 4) {
        const int kk = k0 + khalf;               // even
        const int pidx = (k0 >> 1) + hi;         // B pair-row index

        v2f a0 = *(const v2f*)&As[arow0 * LDA + kk];
        v2f a1 = *(const v2f*)&As[arow1 * LDA + kk];

        const float* bp = &Bs2[pidx * LDBP + 2 * nloc];
        v2f b0 = *(const v2f*)(bp +  0);
        v2f b1 = *(const v2f*)(bp + 32);
        v2f b2 = *(const v2f*)(bp + 64);
        v2f b3 = *(const v2f*)(bp + 96);

        WMMA(acc00, a0, b0);
        WMMA(acc01, a0, b1);
        WMMA(acc02, a0, b2);
        WMMA(acc03, a0, b3);
        WMMA(acc10, a1, b0);
        WMMA(acc11, a1, b1);
        WMMA(acc12, a1, b2);
        WMMA(acc13, a1, b3);
    }

    // ---- Epilogue: scale by inv_norm[col], store ----
    // C/D layout: VGPR v holds row M=v (lanes 0-15) and M=v+8 (lanes 16-31).
    const int grow0 = m0 + wm * 32 + hi * 8;     // M-subtile 0 base row
    const int grow1 = grow0 + 16;                // M-subtile 1 base row
    const int gcol0 = n0 + wn * 64 + l15;

    float sc0 = inv_norm[gcol0];
    float sc1 = inv_norm[gcol0 + 16];
    float sc2 = inv_norm[gcol0 + 32];
    float sc3 = inv_norm[gcol0 + 48];

#pragma unroll
    for (int v = 0; v < 8; ++v) {
        size_t r0 = (size_t)(grow0 + v) * UNITS;
        out[r0 + gcol0]      = acc00[v] * sc0;
        out[r0 + gcol0 + 16] = acc01[v] * sc1;
        out[r0 + gcol0 + 32] = acc02[v] * sc2;
        out[r0 + gcol0 + 48] = acc03[v] * sc3;
        size_t r1 = (size_t)(grow1 + v) * UNITS;
        out[r1 + gcol0]      = acc10[v] * sc0;
        out[r1 + gcol0 + 16] = acc11[v] * sc1;
        out[r1 + gcol0 + 32] = acc12[v] * sc2;
        out[r1 + gcol0 + 48] = acc13[v] * sc3;
    }
}

// ---------------------------------------------------------------------------
// Kernel 3: per-row margin fix-up at (row, label).
// One wave per row: float4 loads + shuffle reduction for ||e||.
// ---------------------------------------------------------------------------
__global__ __launch_bounds__(256) void sphereface_fixup_kernel(
        const float* __restrict__ E,
        const int* __restrict__ labels,
        float* __restrict__ out)
{
    const int wid  = threadIdx.x >> 5;
    const int lane = threadIdx.x & 31;
    const int row  = blockIdx.x * 8 + wid;

    const float4* E4 = (const float4*)E;            // 32 float4 per row
    float4 e = E4[(size_t)row * 32 + lane];
    float s = e.x * e.x + e.y * e.y + e.z * e.z + e.w * e.w;
#pragma unroll
    for (int off = 16; off > 0; off >>= 1)
        s += __shfl_xor(s, off, 32);

    if (lane == 0) {
        float nrm = sqrtf(s);
        int label = labels[row];
        size_t idx = (size_t)row * UNITS + label;
        float sel  = out[idx];                       // already inv_norm-scaled logit
        float ct   = sel / nrm;
        float c2   = ct * ct;
        float c4   = c2 * c2;
        float sign0 = (float)((ct > 0.0f) - (ct < 0.0f));
        float t     = 2.0f * c2 - 1.0f;
        float sign3 = (float)((t > 0.0f) - (t < 0.0f)) * sign0;
        float sign4 = 2.0f * sign0 + sign3 - 3.0f;
        float res   = sign3 * (8.0f * c4 - 8.0f * c2 + 1.0f) + sign4;
        out[idx] = res * nrm;                        // f=1, ff=0 -> replace entry
    }
}

// ---------------------------------------------------------------------------
extern "C" void kernel_launch(void* const* d_in, const int* in_sizes, int n_in,
                              void* d_out, int out_size, void* d_ws, size_t ws_size,
                              hipStream_t stream) {
    const float* emb    = (const float*)d_in[0];   // [16384, 128]
    const int*   labels = (const int*)d_in[1];     // [16384]
    const float* wts    = (const float*)d_in[2];   // [128, 8192]
    float* out = (float*)d_out;                    // [16384, 8192]
    float* inv_norm = (float*)d_ws;                // 8192 floats scratch

    colnorm_kernel<<<UNITS / 256, 256, 0, stream>>>(wts, inv_norm);

    dim3 grid(UNITS / BLOCK_N, NROWS / BLOCK_M);   // 64 x 128
    sphereface_gemm_kernel<<<grid, 256, 0, stream>>>(emb, wts, inv_norm, out);

    sphereface_fixup_kernel<<<NROWS / 8, 256, 0, stream>>>(emb, labels, out);
}